// RBF_49572512530486
// MI455X (gfx1250) — compile-verified
//
#include <hip/hip_runtime.h>
#include <hip/hip_bf16.h>

typedef __attribute__((ext_vector_type(2))) float v2f;
typedef __attribute__((ext_vector_type(8))) float v8f;

#define IN_DIM 64
// Main kernel: block = 256 threads = 8 waves, arranged 4 (M) x 2 (N).
// Each wave: 32 rows x 64 cols = 2 M-subtiles x 4 N-tiles of 16x16.
// Block tile: 128 x 128. Grid: (OUT/128, B/128).

// ---- Pre-pass 1: XS[r] = ||x_r||^2 --------------------------------------
__global__ __launch_bounds__(256) void row_norms_kernel(
    const float* __restrict__ X, float* __restrict__ XS, int n)
{
    int t = blockIdx.x * 256 + threadIdx.x;
    if (t >= n) return;
    const float4* p = (const float4*)(X + (size_t)t * IN_DIM);
    float s = 0.0f;
#pragma unroll
    for (int i = 0; i < IN_DIM / 4; ++i) {
        float4 v = p[i];
        s += v.x * v.x + v.y * v.y + v.z * v.z + v.w * v.w;
    }
    XS[t] = s;
}

// ---- Pre-pass 2: IS2[c] = exp(-2*ls), CSIS[c] = ||c||^2 * IS2[c] --------
__global__ __launch_bounds__(256) void col_norms_kernel(
    const float* __restrict__ C, const float* __restrict__ LS,
    float* __restrict__ IS2, float* __restrict__ CSIS, int n)
{
    int t = blockIdx.x * 256 + threadIdx.x;
    if (t >= n) return;
    const float4* p = (const float4*)(C + (size_t)t * IN_DIM);
    float s = 0.0f;
#pragma unroll
    for (int i = 0; i < IN_DIM / 4; ++i) {
        float4 v = p[i];
        s += v.x * v.x + v.y * v.y + v.z * v.z + v.w * v.w;
    }
    float is2 = __expf(-2.0f * LS[t]);
    IS2[t]  = is2;
    CSIS[t] = s * is2;
}

// ---- Main GEMM + epilogue on V_WMMA_F32_16X16X4_F32 ---------------------
// out = min(exp(2*is2*dot - is2*||x||^2 - is2*||c||^2), 1)
__global__ __launch_bounds__(256) void rbf_wmma_kernel(
    const float* __restrict__ X,     // [B, 64]
    const float* __restrict__ C,     // [OUT, 64]
    const float* __restrict__ XS,    // [B]
    const float* __restrict__ IS2,   // [OUT]
    const float* __restrict__ CSIS,  // [OUT]
    float* __restrict__ out,         // [B, OUT]
    int OUT_DIM)
{
    const int wave = threadIdx.x >> 5;   // 0..7
    const int lane = threadIdx.x & 31;
    const int half = lane >> 4;
    const int lr   = lane & 15;
    const int mwave = wave & 3;          // 0..3 along M
    const int nwave = wave >> 2;         // 0..1 along N

    const int rowBase = blockIdx.y * 128 + mwave * 32;
    const int colBase = blockIdx.x * 128 + nwave * 64;

    // A fragment addressing: half 0 -> K = {k,k+1}, half 1 -> K = {k+2,k+3}
    const float* xrow0 = X + (size_t)(rowBase + lr) * IN_DIM + 2 * half;
    const float* xrow1 = xrow0 + 16 * IN_DIM;
    const float* crow  = C + (size_t)(colBase + lr) * IN_DIM + 2 * half;

    v8f acc[2][4] = {};   // [M-subtile][N-tile]

#pragma unroll
    for (int k = 0; k < IN_DIM; k += 4) {
        v2f a0 = *(const v2f*)(xrow0 + k);
        v2f a1 = *(const v2f*)(xrow1 + k);
        v2f b0 = *(const v2f*)(crow + k);
        v2f b1 = *(const v2f*)(crow + 16 * IN_DIM + k);
        v2f b2 = *(const v2f*)(crow + 32 * IN_DIM + k);
        v2f b3 = *(const v2f*)(crow + 48 * IN_DIM + k);

        acc[0][0] = __builtin_amdgcn_wmma_f32_16x16x4_f32(false, a0, false, b0,
                                                          (short)0, acc[0][0], false, false);
        acc[0][1] = __builtin_amdgcn_wmma_f32_16x16x4_f32(false, a0, false, b1,
                                                          (short)0, acc[0][1], false, false);
        acc[0][2] = __builtin_amdgcn_wmma_f32_16x16x4_f32(false, a0, false, b2,
                                                          (short)0, acc[0][2], false, false);
        acc[0][3] = __builtin_amdgcn_wmma_f32_16x16x4_f32(false, a0, false, b3,
                                                          (short)0, acc[0][3], false, false);
        acc[1][0] = __builtin_amdgcn_wmma_f32_16x16x4_f32(false, a1, false, b0,
                                                          (short)0, acc[1][0], false, false);
        acc[1][1] = __builtin_amdgcn_wmma_f32_16x16x4_f32(false, a1, false, b1,
                                                          (short)0, acc[1][1], false, false);
        acc[1][2] = __builtin_amdgcn_wmma_f32_16x16x4_f32(false, a1, false, b2,
                                                          (short)0, acc[1][2], false, false);
        acc[1][3] = __builtin_amdgcn_wmma_f32_16x16x4_f32(false, a1, false, b3,
                                                          (short)0, acc[1][3], false, false);
    }

    // Per-lane epilogue constants (precomputed by the pre-pass kernels)
    const float xv0 = XS[rowBase + lr];          // ||x||^2 for row lr, subtile 0
    const float xv1 = XS[rowBase + 16 + lr];     // subtile 1
    float is2v[4], csv[4], t2v[4];
#pragma unroll
    for (int j = 0; j < 4; ++j) {
        const int col = colBase + 16 * j + lr;
        is2v[j] = IS2[col];
        csv[j]  = CSIS[col];
        t2v[j]  = 2.0f * is2v[j];
    }

    // D layout: VGPR r -> lanes 0-15: (M=r, N=lane); lanes 16-31: (M=r+8, N=lane-16)
#pragma unroll
    for (int mt = 0; mt < 2; ++mt) {
        const float xv = mt ? xv1 : xv0;
        const int rowOff = rowBase + mt * 16;
#pragma unroll
        for (int j = 0; j < 4; ++j) {
            const int col = colBase + 16 * j + lr;
#pragma unroll
            for (int r = 0; r < 8; ++r) {
                const int m = r + 8 * half;
                const float xsm = __shfl(xv, m, 32);
                const float arg = fmaf(acc[mt][j][r], t2v[j],
                                       -fmaf(xsm, is2v[j], csv[j]));
                const float val = fminf(__expf(arg), 1.0f);
                __builtin_nontemporal_store(
                    val, out + (size_t)(rowOff + m) * OUT_DIM + col);
            }
        }
    }
}

extern "C" void kernel_launch(void* const* d_in, const int* in_sizes, int n_in,
                              void* d_out, int out_size, void* d_ws, size_t ws_size,
                              hipStream_t stream) {
    const float* X  = (const float*)d_in[0];  // [B, 64]
    const float* C  = (const float*)d_in[1];  // [OUT, 64]
    const float* LS = (const float*)d_in[2];  // [OUT]
    float* out = (float*)d_out;

    const int B   = in_sizes[0] / IN_DIM;   // 16384
    const int OUT = in_sizes[2];            // 4096

    float* XS   = (float*)d_ws;             // [B]
    float* IS2  = XS + B;                   // [OUT]
    float* CSIS = IS2 + OUT;                // [OUT]

    row_norms_kernel<<<(B + 255) / 256, 256, 0, stream>>>(X, XS, B);
    col_norms_kernel<<<(OUT + 255) / 256, 256, 0, stream>>>(C, LS, IS2, CSIS, OUT);

    dim3 grid(OUT / 128, B / 128);
    rbf_wmma_kernel<<<grid, 256, 0, stream>>>(X, C, XS, IS2, CSIS, out, OUT);
}